// BlockDP_52407190946258
// MI455X (gfx1250) — compile-verified
//
#include <hip/hip_runtime.h>
#include <hip/hip_bf16.h>
#include <math.h>

// ---------------------------------------------------------------------------
// Problem constants (from reference setup_inputs)
// ---------------------------------------------------------------------------
#define BB    16
#define TT    8192
#define CF    256          // feature channels after GLU
#define PP    32           // predictor channels
#define LMAX  (TT + 2)     // 8194
#define LPAD  8196         // padded to multiple of 3

// Output layout (floats), concatenated in reference return order
#define OUT_XEVS    ((size_t)0)
#define OUT_LENS    ((size_t)BB * CF * LPAD)                 // 33,570,816
#define OUT_BMOVES  (OUT_LENS + BB)
#define OUT_WEIGHTS (OUT_BMOVES + (size_t)BB * TT)

// Workspace layout (floats)
#define WS_FEAT   ((size_t)0)                                // [B][T][C] 128 MB
#define WS_J1     ((size_t)BB * TT * CF)                     // [B][P][T] 16 MB
#define WS_J2     (WS_J1 + (size_t)BB * PP * TT)             // [B][P][T] 16 MB
#define WS_MOVES  (WS_J2 + (size_t)BB * PP * TT)             // [B][T]
#define WS_POSES  (WS_MOVES + (size_t)BB * TT)               // [B][T]

typedef __attribute__((ext_vector_type(2))) float v2f;
typedef __attribute__((ext_vector_type(8))) float v8f;

__device__ __forceinline__ float sigmoidf(float v) {
    return 1.0f / (1.0f + __expf(-v));
}

// ---------------------------------------------------------------------------
// Kernel 0: zero d_out (harness poisons it; pool uses atomics; pad cols = 0)
// ---------------------------------------------------------------------------
__global__ void k_zero(float* __restrict__ p, size_t n) {
    size_t i = (size_t)blockIdx.x * blockDim.x + threadIdx.x;
    size_t stride = (size_t)gridDim.x * blockDim.x;
    for (; i < n; i += stride) p[i] = 0.0f;
}

// ---------------------------------------------------------------------------
// Kernel 1: main conv (1->512, k=9) + BN(1e-3) + GLU -> features [B][T][C]
// One thread per channel c, 64 time steps per block. x window in LDS.
// ---------------------------------------------------------------------------
__global__ __launch_bounds__(256) void k_main_conv_glu(
    const float* __restrict__ x, const float* __restrict__ cw,
    const float* __restrict__ bn_g, const float* __restrict__ bn_b,
    const float* __restrict__ bn_m, const float* __restrict__ bn_v,
    float* __restrict__ feat) {
    const int b  = blockIdx.x >> 7;            // 128 t-blocks per batch
    const int tb = (blockIdx.x & 127) * 64;
    const int c  = threadIdx.x;                // 0..255
    __shared__ float xs[72];
    for (int i = threadIdx.x; i < 72; i += 256) {
        int t = tb - 4 + i;
        xs[i] = (t >= 0 && t < TT) ? x[(size_t)b * TT + t] : 0.0f;
    }
    __syncthreads();
    float wa[9], wg[9];
    #pragma unroll
    for (int k = 0; k < 9; ++k) {
        wa[k] = cw[c * 9 + k];
        wg[k] = cw[(c + CF) * 9 + k];
    }
    const float sA = bn_g[c]      * rsqrtf(bn_v[c]      + 1e-3f);
    const float bA = bn_b[c]      - bn_m[c]      * sA;
    const float sG = bn_g[c + CF] * rsqrtf(bn_v[c + CF] + 1e-3f);
    const float bG = bn_b[c + CF] - bn_m[c + CF] * sG;
    for (int tt = 0; tt < 64; ++tt) {
        float a = 0.0f, g = 0.0f;
        #pragma unroll
        for (int k = 0; k < 9; ++k) {
            float xv = xs[tt + k];
            a = fmaf(wa[k], xv, a);
            g = fmaf(wg[k], xv, g);
        }
        a = a * sA + bA;
        g = g * sG + bG;
        feat[((size_t)b * TT + tb + tt) * CF + c] = a * sigmoidf(g);
    }
}

// ---------------------------------------------------------------------------
// Kernel 2: predictor conv1 (cat[x,x^2] 2->32, k=31) + BN(1e-5) + swish
// -> j1 [B][P][T].  One thread per time step, loop over 32 out channels.
// ---------------------------------------------------------------------------
__global__ __launch_bounds__(256) void k_p1(
    const float* __restrict__ x, const float* __restrict__ w1,
    const float* __restrict__ b1,
    const float* __restrict__ g1, const float* __restrict__ bb1,
    const float* __restrict__ m1, const float* __restrict__ v1,
    float* __restrict__ j1) {
    const int b  = blockIdx.x >> 5;            // 32 t-blocks per batch
    const int tb = (blockIdx.x & 31) * 256;
    __shared__ float xs[286];                  // 256 + 30 halo
    __shared__ float lw[PP * 62];              // p1_w flat copy [co][ci][k]
    for (int i = threadIdx.x; i < 286; i += 256) {
        int t = tb - 15 + i;
        xs[i] = (t >= 0 && t < TT) ? x[(size_t)b * TT + t] : 0.0f;
    }
    for (int i = threadIdx.x; i < PP * 62; i += 256) lw[i] = w1[i];
    __syncthreads();
    const int tt = threadIdx.x;
    for (int co = 0; co < PP; ++co) {
        float acc = b1[co];
        #pragma unroll
        for (int k = 0; k < 31; ++k) {
            float xv = xs[tt + k];
            acc = fmaf(lw[co * 62 + k],      xv,      acc);
            acc = fmaf(lw[co * 62 + 31 + k], xv * xv, acc);
        }
        float s = g1[co] * rsqrtf(v1[co] + 1e-5f);
        acc = acc * s + (bb1[co] - m1[co] * s);
        acc = acc * sigmoidf(acc);
        j1[((size_t)(b * PP + co)) * TT + tb + tt] = acc;
    }
}

// ---------------------------------------------------------------------------
// Kernel 3: predictor conv2 (32->32, k=15) as GEMM via V_WMMA_F32_16X16X4_F32
// GEMM: D[co, t] = sum_{ci,k} W[co, ci*16+k] * j1[ci, t+k-7]   (k=15 padded
// to 16 per ci so K-chunks of 4 never straddle a tap boundary; K_total=512).
// Weights zero-padded into LDS (row stride 516 -> conflict-free A reads) so
// the inner loop is branch-free: ds_load + v_wmma only.
// 8 waves/block: 2 co-tiles x 4 t-tiles (64 time steps).
// ---------------------------------------------------------------------------
#if __has_builtin(__builtin_amdgcn_wmma_f32_16x16x4_f32)
#define USE_WMMA_F32 1
#else
#define USE_WMMA_F32 0
#endif

#define WSTRIDE 516   // 512 padded-K + 4 (stride mod 64 == 4 -> no bank dup)

__global__ __launch_bounds__(256) void k_p2_wmma(
    const float* __restrict__ j1, const float* __restrict__ w2,
    const float* __restrict__ b2,
    const float* __restrict__ g2, const float* __restrict__ bb2,
    const float* __restrict__ m2, const float* __restrict__ v2,
    float* __restrict__ j2) {
    const int b     = blockIdx.x >> 7;         // 128 blocks per batch (T/64)
    const int tbase = (blockIdx.x & 127) * 64;
    __shared__ float wpad[PP * WSTRIDE];       // [co][512], k=15 slots = 0
    __shared__ float lj[PP][80];               // [ci][full 80 cols, finite]
    // Zero-padded weights: u = ci*16 + k, k==15 -> 0.
    for (int i = threadIdx.x; i < PP * 512; i += 256) {
        const int co = i >> 9, u = i & 511;
        const int ci = u >> 4, k = u & 15;
        wpad[co * WSTRIDE + u] = (k < 15) ? w2[(co * PP + ci) * 15 + k] : 0.0f;
    }
    // Full 80-wide halo fill so every B read is finite (k=15 column is a
    // don't-care multiplied by the zero weight).
    for (int i = threadIdx.x; i < PP * 80; i += 256) {
        const int ci = i / 80, xo = i % 80;
        const int t = tbase - 7 + xo;
        lj[ci][xo] = (t >= 0 && t < TT) ? j1[((size_t)(b * PP + ci)) * TT + t]
                                        : 0.0f;
    }
    __syncthreads();
    const int wave = threadIdx.x >> 5;
    const int lane = threadIdx.x & 31;
    const int half = lane >> 4;
    const int ln   = lane & 15;
    const int wt   = wave & 3;                 // time tile within block
    const int co0  = (wave >> 2) * 16;         // 0 or 16

#if USE_WMMA_F32
    // A layout (16x4 f32): VGPR0 holds K=2*half, VGPR1 holds K=2*half+1.
    const float* __restrict__ aw = &wpad[(co0 + ln) * WSTRIDE];
    const int bbase = wt * 16 + ln;
    v8f acc = {};
    #pragma unroll 8
    for (int u = 0; u < 512; u += 4) {
        const int ci = u >> 4;
        const int kl = (u & 15) + half * 2;    // tap index for this lane-half
        v2f a, bo;
        a.x  = aw[u + half * 2];
        a.y  = aw[u + half * 2 + 1];
        bo.x = lj[ci][bbase + kl];
        bo.y = lj[ci][bbase + kl + 1];
        acc = __builtin_amdgcn_wmma_f32_16x16x4_f32(
            /*neg_a=*/false, a, /*neg_b=*/false, bo,
            /*c_mod=*/(short)0, acc, /*reuse_a=*/false, /*reuse_b=*/false);
    }
    // Epilogue: lane<16 owns rows M=r, lane>=16 owns rows M=r+8; N = ln.
    const int n = tbase + wt * 16 + ln;
    #pragma unroll
    for (int r = 0; r < 8; ++r) {
        const int m = co0 + r + 8 * half;
        float s   = g2[m] * rsqrtf(v2[m] + 1e-5f);
        float val = acc[r] + b2[m];
        val = val * s + (bb2[m] - m2[m] * s);
        val = val * sigmoidf(val);
        j2[((size_t)(b * PP + m)) * TT + n] = val;
    }
#else
    // VALU fallback: each lane computes 2 (co,t) outputs directly.
    for (int r = 0; r < 2; ++r) {
        const int m = co0 + half * 8 + r * 4 + (ln >> 2);
        const int n = tbase + wt * 16 + (ln & 3) * 4;
        float val = b2[m];
        for (int ci = 0; ci < PP; ++ci)
            for (int k = 0; k < 15; ++k)
                val = fmaf(wpad[m * WSTRIDE + ci * 16 + k],
                           lj[ci][(n - tbase) + k], val);
        float s = g2[m] * rsqrtf(v2[m] + 1e-5f);
        val = val * s + (bb2[m] - m2[m] * s);
        val = val * sigmoidf(val);
        j2[((size_t)(b * PP + m)) * TT + n] = val;
    }
#endif
}

// ---------------------------------------------------------------------------
// Kernel 4: predictor conv3 (32->2, k=15) + sigmoids -> weights, bmoves (out)
// and moves = sigmoid * norm_mean (ws).
// ---------------------------------------------------------------------------
__global__ __launch_bounds__(256) void k_p3(
    const float* __restrict__ j2, const float* __restrict__ w3,
    const float* __restrict__ b3, const float* __restrict__ nm,
    float* __restrict__ out, float* __restrict__ moves_ws) {
    const int b  = blockIdx.x >> 5;            // 32 t-blocks per batch
    const int tb = (blockIdx.x & 31) * 256;
    __shared__ float lj[PP][272];              // 256 + 14 halo, padded stride
    __shared__ float lw[2 * PP * 15];
    for (int i = threadIdx.x; i < PP * 270; i += 256) {
        int ci = i / 270, xo = i % 270;
        int t = tb - 7 + xo;
        lj[ci][xo] = (t >= 0 && t < TT) ? j2[((size_t)(b * PP + ci)) * TT + t]
                                        : 0.0f;
    }
    for (int i = threadIdx.x; i < 2 * PP * 15; i += 256) lw[i] = w3[i];
    __syncthreads();
    const int tt = threadIdx.x;
    float acc0 = b3[0], acc1 = b3[1];
    for (int ci = 0; ci < PP; ++ci) {
        #pragma unroll
        for (int k = 0; k < 15; ++k) {
            float xv = lj[ci][tt + k];
            acc0 = fmaf(lw[ci * 15 + k],            xv, acc0);
            acc1 = fmaf(lw[PP * 15 + ci * 15 + k],  xv, acc1);
        }
    }
    const size_t idx = (size_t)b * TT + tb + tt;
    float wgt = sigmoidf(acc0);
    float mv  = sigmoidf(acc1);
    out[OUT_WEIGHTS + idx] = wgt;
    out[OUT_BMOVES  + idx] = mv;                // bmoves (pre-renorm)
    moves_ws[idx] = mv * nm[0];                 // moves * norm_mean
}

// ---------------------------------------------------------------------------
// Kernel 5: per-batch inclusive cumsum of moves -> poses; lens = floor+2.
// One block per batch, chunked 256-wide Hillis-Steele scan with carry.
// ---------------------------------------------------------------------------
__global__ __launch_bounds__(256) void k_scan(
    const float* __restrict__ moves, float* __restrict__ poses,
    float* __restrict__ out_lens) {
    const int b = blockIdx.x;
    __shared__ float s[256];
    float carry = 0.0f;
    for (int ch = 0; ch < TT / 256; ++ch) {
        const int t = ch * 256 + threadIdx.x;
        s[threadIdx.x] = moves[(size_t)b * TT + t];
        __syncthreads();
        for (int off = 1; off < 256; off <<= 1) {
            float add = (threadIdx.x >= (unsigned)off) ? s[threadIdx.x - off]
                                                       : 0.0f;
            __syncthreads();
            s[threadIdx.x] += add;
            __syncthreads();
        }
        poses[(size_t)b * TT + t] = carry + s[threadIdx.x];
        carry += s[255];
        __syncthreads();
    }
    if (threadIdx.x == 0) out_lens[b] = floorf(carry) + 2.0f;
}

// ---------------------------------------------------------------------------
// Kernel 6: CIF pooling.  One block = (batch, 256-step time segment); one
// thread per channel.  moves<1 => floor(poses) advances by <=1, so each
// thread register-accumulates the current (bin, bin+1) pair and flushes with
// a float atomic only when the bin advances (segment boundaries overlap, so
// flushes must be atomic).
// ---------------------------------------------------------------------------
__global__ __launch_bounds__(256) void k_pool(
    const float* __restrict__ feat, const float* __restrict__ poses,
    const float* __restrict__ wgt, float* __restrict__ out) {
    const int b   = blockIdx.x >> 5;           // 32 segments per batch
    const int seg = blockIdx.x & 31;
    const int c   = threadIdx.x;
    const int ts  = seg * 256, te = ts + 256;
    float* const base = out + ((size_t)b * CF + c) * LPAD;
    float acc0 = 0.0f, acc1 = 0.0f;
    int cur = -1;
    for (int t = ts; t < te; ++t) {
        const size_t it = (size_t)b * TT + t;
        const float p  = poses[it];
        const float fw = feat[it * CF + c] * wgt[it];
        const float fl = floorf(p);
        const int bin  = (int)fl;
        const float w2 = p - fl;
        const float w1 = 1.0f - w2;
        if (bin != cur) {
            if (cur >= 0) {
                atomicAdd(base + cur, acc0);
                if (bin == cur + 1) {
                    acc0 = acc1; acc1 = 0.0f;
                } else {                       // jump >= 2 (safety)
                    atomicAdd(base + cur + 1, acc1);
                    acc0 = 0.0f; acc1 = 0.0f;
                }
            }
            cur = bin;
        }
        acc0 = fmaf(w1, fw, acc0);
        acc1 = fmaf(w2, fw, acc1);
    }
    if (cur >= 0) {
        atomicAdd(base + cur,     acc0);
        atomicAdd(base + cur + 1, acc1);
    }
}

// ---------------------------------------------------------------------------
// Host launcher
// ---------------------------------------------------------------------------
extern "C" void kernel_launch(void* const* d_in, const int* in_sizes, int n_in,
                              void* d_out, int out_size, void* d_ws,
                              size_t ws_size, hipStream_t stream) {
    const float* x      = (const float*)d_in[0];
    const float* conv_w = (const float*)d_in[1];
    const float* bn_g   = (const float*)d_in[2];
    const float* bn_b   = (const float*)d_in[3];
    const float* bn_m   = (const float*)d_in[4];
    const float* bn_v   = (const float*)d_in[5];
    const float* p1_w   = (const float*)d_in[6];
    const float* p1_b   = (const float*)d_in[7];
    const float* pbn1_g = (const float*)d_in[8];
    const float* pbn1_b = (const float*)d_in[9];
    const float* pbn1_m = (const float*)d_in[10];
    const float* pbn1_v = (const float*)d_in[11];
    const float* p2_w   = (const float*)d_in[12];
    const float* p2_b   = (const float*)d_in[13];
    const float* pbn2_g = (const float*)d_in[14];
    const float* pbn2_b = (const float*)d_in[15];
    const float* pbn2_m = (const float*)d_in[16];
    const float* pbn2_v = (const float*)d_in[17];
    const float* p3_w   = (const float*)d_in[18];
    const float* p3_b   = (const float*)d_in[19];
    const float* nmean  = (const float*)d_in[20];

    float* out = (float*)d_out;
    float* ws  = (float*)d_ws;
    float* feat  = ws + WS_FEAT;
    float* j1    = ws + WS_J1;
    float* j2    = ws + WS_J2;
    float* moves = ws + WS_MOVES;
    float* poses = ws + WS_POSES;

    // 0) zero the whole output (pool accumulates with atomics; pad cols = 0)
    k_zero<<<4096, 256, 0, stream>>>(out, (size_t)out_size);

    // 1) main conv + BN + GLU -> features [B][T][C]
    k_main_conv_glu<<<BB * (TT / 64), 256, 0, stream>>>(
        x, conv_w, bn_g, bn_b, bn_m, bn_v, feat);

    // 2) predictor conv1 + BN + swish -> j1
    k_p1<<<BB * (TT / 256), 256, 0, stream>>>(
        x, p1_w, p1_b, pbn1_g, pbn1_b, pbn1_m, pbn1_v, j1);

    // 3) predictor conv2 (WMMA f32) + BN + swish -> j2
    k_p2_wmma<<<BB * (TT / 64), 256, 0, stream>>>(
        j1, p2_w, p2_b, pbn2_g, pbn2_b, pbn2_m, pbn2_v, j2);

    // 4) predictor conv3 + sigmoids -> weights/bmoves (out), moves (ws)
    k_p3<<<BB * (TT / 256), 256, 0, stream>>>(
        j2, p3_w, p3_b, nmean, out, moves);

    // 5) cumsum -> poses; lens
    k_scan<<<BB, 256, 0, stream>>>(moves, poses, out + OUT_LENS);

    // 6) CIF pooling -> x_evs
    k_pool<<<BB * 32, 256, 0, stream>>>(feat, poses, out + OUT_WEIGHTS, out);
}